// STGCNBlock_7447473291365
// MI455X (gfx1250) — compile-verified
//
#include <hip/hip_runtime.h>

// ---------------------------------------------------------------------------
// STGCN block for MI455X (gfx1250, wave32).
//  - f32 WMMA (V_WMMA_F32_16X16X4_F32) for both conv1d-as-GEMM kernels and the
//    dual GEMM (xl = xn@W_l, xr = xn@W_r): exact f32 precision on the matrix pipe.
//  - Conv input tiles staged to LDS with GLOBAL_LOAD_ASYNC_TO_LDS_B128
//    (ASYNCcnt-tracked, no VGPR round trip) for interior tiles; predicated
//    scalar staging only on the zero-padded boundary tiles.
//  - GATv2 edge softmax: 3 passes, wave-per-edge, L2-resident gathers,
//    order-preserving uint encoding for float atomicMax.
// ---------------------------------------------------------------------------

#define B_  64
#define C_  64
#define T_  2048
#define D_  64
#define K_  9
#define N_  (B_ * T_)          // 131072 nodes
#define E_  2000000
#define EN_ (E_ + N_)          // edges incl. self loops = 2131072
#define EPS_ 1e-5f

typedef __attribute__((ext_vector_type(2))) float v2f;
typedef __attribute__((ext_vector_type(8))) float v8f;

// D(16x16,f32) = A(16x4,f32) x B(4x16,f32) + C
// Assumed lane layout (per CDNA5 ISA 7.12.2):
//   A: VGPR0 = {K=0 | K=2}, VGPR1 = {K=1 | K=3}; M = lane&15
//   B: mirrored K split; N = lane&15
//   C/D: VGPR v: lanes0-15 -> M=v, lanes16-31 -> M=v+8; N = lane&15
__device__ __forceinline__ v8f wmma_f32x4(v2f a, v2f b, v8f c) {
  return __builtin_amdgcn_wmma_f32_16x16x4_f32(false, a, false, b,
                                               (short)0, c, false, false);
}

// Async global->LDS copy, 16B per active lane (ASYNCcnt tracked).
// %0 = LDS byte address (low 32 bits of generic shared pointer),
// %1 = 64-bit global address.
__device__ __forceinline__ void async_copy_b128(unsigned lds_off, const void* gptr) {
  asm volatile("global_load_async_to_lds_b128 %0, %1, off"
               :: "v"(lds_off), "v"(gptr) : "memory");
}
__device__ __forceinline__ void wait_async0() {
  asm volatile("s_wait_asynccnt 0x0" ::: "memory");
}

// ------------------------------ elementwise --------------------------------

__global__ void bn0_kernel(const float* __restrict__ x, float* __restrict__ xbn,
                           const float* __restrict__ g, const float* __restrict__ b,
                           const float* __restrict__ mn, const float* __restrict__ vr) {
  int idx = blockIdx.x * blockDim.x + threadIdx.x;
  if (idx >= B_ * C_ * T_) return;
  int c = (idx >> 11) & 63;                     // idx = b*C*T + c*T + t, T=2048
  float inv = g[c] * rsqrtf(vr[c] + EPS_);
  xbn[idx] = (x[idx] - mn[c]) * inv + b[c];
}

// W[d, c, k] (D x C x K) -> Wrep[k][d][c]  (A-fragment friendly)
__global__ void repack_kernel(const float* __restrict__ w, float* __restrict__ wr) {
  int idx = blockIdx.x * blockDim.x + threadIdx.x;
  if (idx >= D_ * C_ * K_) return;
  int d = idx / (C_ * K_);
  int r = idx - d * (C_ * K_);
  int c = r / K_;
  int k = r - c * K_;
  wr[k * (D_ * C_) + d * C_ + c] = w[idx];
}

__global__ void zero_kernel(float* __restrict__ p, int n) {
  int i = blockIdx.x * blockDim.x + threadIdx.x;
  if (i < n) p[i] = 0.f;
}

// post-GAT: x = relu(bn1(xgat + bias_gat)), in place
__global__ void gat_bn_relu_kernel(float* __restrict__ xgat,
                                   const float* __restrict__ bg,
                                   const float* __restrict__ g1, const float* __restrict__ b1,
                                   const float* __restrict__ m1, const float* __restrict__ v1) {
  int idx = blockIdx.x * blockDim.x + threadIdx.x;
  if (idx >= N_ * D_) return;
  int d = idx & 63;
  float val = xgat[idx] + bg[d];
  float inv = g1[d] * rsqrtf(v1[d] + EPS_);
  val = (val - m1[d]) * inv + b1[d];
  xgat[idx] = fmaxf(val, 0.f);
}

// ------------------------------ dual GEMM ----------------------------------
// xn [131072 x 64] (raw view of xbn) times W_l and W_r [64 x 64].
// 256 thr = 8 waves; wave handles 16 rows x 64 cols for BOTH outputs.
__global__ void gemm_xlxr_kernel(const float* __restrict__ xn,
                                 const float* __restrict__ wl,
                                 const float* __restrict__ wr,
                                 float* __restrict__ xl, float* __restrict__ xr) {
  const int lane  = threadIdx.x & 31;
  const int wave  = threadIdx.x >> 5;
  const int m0    = blockIdx.x * 128 + wave * 16;
  const int l15   = lane & 15;
  const int khalf = (lane >> 4) << 1;

  const v8f zero8 = {0.f, 0.f, 0.f, 0.f, 0.f, 0.f, 0.f, 0.f};
  v8f accL[4], accR[4];
#pragma unroll
  for (int i = 0; i < 4; ++i) { accL[i] = zero8; accR[i] = zero8; }

  for (int cb = 0; cb < 16; ++cb) {
    const float* ap = xn + (m0 + l15) * 64 + cb * 4 + khalf;
    v2f a; a.x = ap[0]; a.y = ap[1];
#pragma unroll
    for (int nt = 0; nt < 4; ++nt) {
      int r0 = (cb * 4 + khalf) * 64 + nt * 16 + l15;
      v2f bl; bl.x = wl[r0]; bl.y = wl[r0 + 64];
      v2f br; br.x = wr[r0]; br.y = wr[r0 + 64];
      accL[nt] = wmma_f32x4(a, bl, accL[nt]);
      accR[nt] = wmma_f32x4(a, br, accR[nt]);
    }
  }

  const int hi8 = (lane >> 4) * 8;
#pragma unroll
  for (int nt = 0; nt < 4; ++nt) {
#pragma unroll
    for (int v = 0; v < 8; ++v) {
      int row = m0 + v + hi8;
      int col = nt * 16 + l15;
      xl[row * 64 + col] = accL[nt][v];
      xr[row * 64 + col] = accR[nt][v];
    }
  }
}

// ------------------------------ conv1d (K=9, SAME) -------------------------
// y[b,d,t] = sum_{k,c} xin[b,c,t+k-4] * W[d,c,k]  via WMMA, M=cout, N=time.
// PHASE 0: out = relu(y + bs)              (residual, overwrite d_out)
// PHASE 1: out += relu(bn1(y + bt))        (final accumulate into d_out)
template <int PHASE>
__global__ void conv_kernel(const float* __restrict__ xin,    // [B,64,T] flat
                            const float* __restrict__ wrep,   // [K][D][C]
                            const float* __restrict__ bias,
                            const float* __restrict__ g1, const float* __restrict__ b1,
                            const float* __restrict__ m1, const float* __restrict__ v1,
                            float* __restrict__ out) {
  __shared__ float lds[64 * 136];                 // 34.8 KB
  const int b   = blockIdx.x >> 4;
  const int t0  = (blockIdx.x & 15) * 128;
  const int tid = threadIdx.x;

  // L2-resident weights are re-read by all 1024 blocks: prefetch the 9 planes.
  if (tid < K_) __builtin_prefetch(wrep + tid * (D_ * C_), 0, 3);

  // stage [64 ch][136 cols]; col j -> global t = t0 + j - 4 (zero padded)
  const bool interior = (t0 >= 4) && (t0 + 132 <= T_);
  if (interior) {
    // async 16B/lane copies: 64 rows x 34 float4 = 2176 transfers
    const float* gbase = xin + (b * 64) * T_ + (t0 - 4);
    for (int i = tid; i < 64 * 34; i += 256) {
      int ch = i / 34;
      int q  = i - ch * 34;
      unsigned loff = (unsigned)(unsigned long long)(&lds[ch * 136 + q * 4]);
      async_copy_b128(loff, gbase + ch * T_ + q * 4);
    }
    wait_async0();
  } else {
    for (int i = tid; i < 64 * 136; i += 256) {
      int ch  = i / 136;
      int col = i - ch * 136;
      int tg  = t0 + col - 4;
      float v = 0.f;
      if (tg >= 0 && tg < T_) v = xin[(b * 64 + ch) * T_ + tg];
      lds[i] = v;
    }
  }
  __syncthreads();

  const int lane  = tid & 31;
  const int wave  = tid >> 5;
  const int l15   = lane & 15;
  const int khalf = (lane >> 4) << 1;
  const int tloc  = wave * 16;                    // this wave's 16 time steps

  const v8f zero8 = {0.f, 0.f, 0.f, 0.f, 0.f, 0.f, 0.f, 0.f};
  v8f acc[4];
#pragma unroll
  for (int i = 0; i < 4; ++i) acc[i] = zero8;

  for (int k = 0; k < K_; ++k) {
    for (int cb = 0; cb < 16; ++cb) {
      int c0 = cb * 4 + khalf;
      v2f bf;                                     // B frag: [c][t] from LDS
      bf.x = lds[c0 * 136 + tloc + l15 + k];
      bf.y = lds[(c0 + 1) * 136 + tloc + l15 + k];
#pragma unroll
      for (int mt = 0; mt < 4; ++mt) {            // A frag: weights [d][c]
        const float* wp = wrep + k * (D_ * C_) + (mt * 16 + l15) * 64 + c0;
        v2f a; a.x = wp[0]; a.y = wp[1];
        acc[mt] = wmma_f32x4(a, bf, acc[mt]);
      }
    }
  }

  const int hi8 = (lane >> 4) * 8;
  const int t   = t0 + tloc + l15;                // N index = time (coalesced)
#pragma unroll
  for (int mt = 0; mt < 4; ++mt) {
#pragma unroll
    for (int v = 0; v < 8; ++v) {
      int d = mt * 16 + v + hi8;
      float val = acc[mt][v] + bias[d];
      int oidx = (b * 64 + d) * T_ + t;
      if (PHASE == 0) {
        out[oidx] = fmaxf(val, 0.f);
      } else {
        float inv = g1[d] * rsqrtf(v1[d] + EPS_);
        val = (val - m1[d]) * inv + b1[d];
        out[oidx] += fmaxf(val, 0.f);
      }
    }
  }
}

// ------------------------------ GATv2 edge stage ---------------------------

__device__ __forceinline__ unsigned fenc(float f) {   // order-preserving
  unsigned u = __float_as_uint(f);
  return (u & 0x80000000u) ? ~u : (u | 0x80000000u);
}
__device__ __forceinline__ float fdec(unsigned e) {
  unsigned u = (e & 0x80000000u) ? (e & 0x7fffffffu) : ~e;
  return __uint_as_float(u);
}

// pass 1: e_i = att . leaky_relu(xl[src]+xr[dst]); segment max via atomicMax
__global__ void edge_e_max_kernel(const int* __restrict__ ei,
                                  const float* __restrict__ xl,
                                  const float* __restrict__ xr,
                                  const float* __restrict__ att,
                                  float* __restrict__ ebuf,
                                  unsigned* __restrict__ menc) {
  int eid = blockIdx.x * 8 + (threadIdx.x >> 5);
  if (eid >= EN_) return;
  int lane = threadIdx.x & 31;
  int src, dst;
  if (eid < E_) { src = ei[eid]; dst = ei[E_ + eid]; }
  else          { src = dst = eid - E_; }
  const float* pl = xl + src * 64;
  const float* pr = xr + dst * 64;
  float e = 0.f;
#pragma unroll
  for (int h = 0; h < 2; ++h) {
    int d = lane + h * 32;
    float v = pl[d] + pr[d];
    v = (v > 0.f) ? v : 0.2f * v;                 // leaky_relu(0.2)
    e += v * att[d];
  }
#pragma unroll
  for (int off = 16; off > 0; off >>= 1) e += __shfl_xor(e, off, 32);
  if (lane == 0) {
    ebuf[eid] = e;
    atomicMax(&menc[dst], fenc(e));
  }
}

// pass 2: a_i = exp(e_i - m[dst]); segment sum
__global__ void edge_exp_sum_kernel(const int* __restrict__ ei,
                                    float* __restrict__ ebuf,
                                    const unsigned* __restrict__ menc,
                                    float* __restrict__ s) {
  int i = blockIdx.x * blockDim.x + threadIdx.x;
  if (i >= EN_) return;
  int dst = (i < E_) ? ei[E_ + i] : (i - E_);
  float a = expf(ebuf[i] - fdec(menc[dst]));
  ebuf[i] = a;
  atomicAdd(&s[dst], a);
}

// pass 3: xgat[dst] += (a_i / s[dst]) * xl[src]
__global__ void edge_scatter_kernel(const int* __restrict__ ei,
                                    const float* __restrict__ ebuf,
                                    const float* __restrict__ s,
                                    const float* __restrict__ xl,
                                    float* __restrict__ xgat) {
  int eid = blockIdx.x * 8 + (threadIdx.x >> 5);
  if (eid >= EN_) return;
  int lane = threadIdx.x & 31;
  int src, dst;
  if (eid < E_) { src = ei[eid]; dst = ei[E_ + eid]; }
  else          { src = dst = eid - E_; }
  float alpha = ebuf[eid] / s[dst];
  const float* pl = xl + src * 64;
  float* po = xgat + dst * 64;
  atomicAdd(&po[lane],      alpha * pl[lane]);
  atomicAdd(&po[lane + 32], alpha * pl[lane + 32]);
}

// ------------------------------ launch -------------------------------------

extern "C" void kernel_launch(void* const* d_in, const int* in_sizes, int n_in,
                              void* d_out, int out_size, void* d_ws, size_t ws_size,
                              hipStream_t stream) {
  (void)in_sizes; (void)n_in; (void)out_size; (void)ws_size;
  const float* x   = (const float*)d_in[0];
  const int*   ei  = (const int*)d_in[1];
  const float* Wl  = (const float*)d_in[2];
  const float* Wr  = (const float*)d_in[3];
  const float* att = (const float*)d_in[4];
  const float* bg  = (const float*)d_in[5];
  const float* g0  = (const float*)d_in[6];
  const float* b0  = (const float*)d_in[7];
  const float* mn0 = (const float*)d_in[8];
  const float* vr0 = (const float*)d_in[9];
  const float* g1  = (const float*)d_in[10];
  const float* b1  = (const float*)d_in[11];
  const float* mn1 = (const float*)d_in[12];
  const float* vr1 = (const float*)d_in[13];
  const float* Wt  = (const float*)d_in[14];
  const float* bt  = (const float*)d_in[15];
  const float* Ws  = (const float*)d_in[16];
  const float* bs  = (const float*)d_in[17];
  float* out = (float*)d_out;
  float* ws  = (float*)d_ws;

  // workspace layout (floats)
  float*    xbn   = ws;                        // 8388608
  float*    xl    = ws + 8388608;              // 8388608
  float*    xr    = ws + 16777216;             // 8388608
  float*    xgat  = ws + 25165824;             // 8388608 (reused as conv2 input)
  unsigned* menc  = (unsigned*)(ws + 33554432);// 131072
  float*    ssum  = ws + 33685504;             // 131072
  float*    ebuf  = ws + 33816576;             // 2131072
  float*    wrepS = ws + 35947648;             // 36864
  float*    wrepT = ws + 35984512;             // 36864

  const int NEL = B_ * C_ * T_;                // 8388608

  bn0_kernel<<<(NEL + 255) / 256, 256, 0, stream>>>(x, xbn, g0, b0, mn0, vr0);

  repack_kernel<<<(D_ * C_ * K_ + 255) / 256, 256, 0, stream>>>(Ws, wrepS);
  repack_kernel<<<(D_ * C_ * K_ + 255) / 256, 256, 0, stream>>>(Wt, wrepT);

  // zero the atomically-accumulated region: xgat | menc | ssum (contiguous)
  zero_kernel<<<(8650752 + 255) / 256, 256, 0, stream>>>(xgat, 8650752);

  // residual = relu(conv1d(xbn, Ws) + bs)  -> d_out
  conv_kernel<0><<<1024, 256, 0, stream>>>(xbn, wrepS, bs, g1, b1, mn1, vr1, out);

  // xl / xr GEMMs
  gemm_xlxr_kernel<<<1024, 256, 0, stream>>>(xbn, Wl, Wr, xl, xr);

  // GATv2 softmax aggregation
  edge_e_max_kernel<<<(EN_ + 7) / 8, 256, 0, stream>>>(ei, xl, xr, att, ebuf, menc);
  edge_exp_sum_kernel<<<(EN_ + 255) / 256, 256, 0, stream>>>(ei, ebuf, menc, ssum);
  edge_scatter_kernel<<<(EN_ + 7) / 8, 256, 0, stream>>>(ei, ebuf, ssum, xl, xgat);

  // relu(bn1(xgat + bias_gat)) in place -> conv2 input (raw [B,D,T] view)
  gat_bn_relu_kernel<<<(N_ * D_ + 255) / 256, 256, 0, stream>>>(xgat, bg, g1, b1, mn1, vr1);

  // out += relu(bn1(conv1d(xgat, Wt) + bt))
  conv_kernel<1><<<1024, 256, 0, stream>>>(xgat, wrepT, bt, g1, b1, mn1, vr1, out);
}